// BruteForceEstimatorFast_13048110645539
// MI455X (gfx1250) — compile-verified
//
#include <hip/hip_runtime.h>
#include <hip/hip_bf16.h>
#include <math.h>

// ---------------------------------------------------------------------------
// CNF forward + exact divergence for B=4096, D=66, H=512 on MI455X (gfx1250).
//
// Closed-form trace:
//   tr(J)[b] = s1[b]^T (W2 .* M) s2[b],  M[l,k] = sum_{j<D} W1[j,l] W3[k,j]
// -> 4 GEMMs + one batch-independent 512x512 matrix instead of 66 JVPs.
// All GEMMs run in full fp32 on the CDNA5 matrix pipe (V_WMMA_F32_16X16X4_F32)
// to keep reference fp32 semantics; the problem is memory-bound (~45MB, fits
// in the 192MB L2), so fp32 matrix math is the right precision/speed point.
//
// GEMM core: 64x64 block tile, 4 waves (2x2), each wave owns a 32x32 tile as
// 2x2 fp32 16x16 accumulators. A tile staged row-major, B tile staged
// TRANSPOSED in LDS so every WMMA fragment is one contiguous ds_load_b64
// (no v_mov fixups); +4 dword row pad (stride 36) -> conflict-free frag reads.
// ---------------------------------------------------------------------------

typedef __attribute__((ext_vector_type(2))) float v2f;
typedef __attribute__((ext_vector_type(8))) float v8f;

#define B_SZ 4096
#define D_SZ 66
#define H_SZ 512

constexpr int BM = 64;        // block tile rows
constexpr int BN = 64;        // block tile cols
constexpr int BK = 32;        // k tile
constexpr int NT = 128;       // 4 waves, 2x2; each wave -> 32x32 output

__device__ __forceinline__ v8f wmma4(v2f a, v2f b, v8f c) {
    return __builtin_amdgcn_wmma_f32_16x16x4_f32(
        false, a, false, b, (short)0, c, false, false);
}

// ---------------------------------------------------------------------------
// C = (W1[:D,:]^T @ W3^T) .* W2      (512x512, K=66; tiny, VALU kernel)
// ---------------------------------------------------------------------------
__global__ __launch_bounds__(256)
void precompute_C(const float* __restrict__ W1, const float* __restrict__ W2,
                  const float* __restrict__ W3, float* __restrict__ Cm)
{
    int idx = blockIdx.x * blockDim.x + threadIdx.x;   // 0 .. 512*512-1
    int k = idx & (H_SZ - 1);
    int l = idx >> 9;
    float m = 0.0f;
    #pragma unroll 2
    for (int j = 0; j < D_SZ; ++j)
        m = fmaf(W1[j * H_SZ + l], W3[k * D_SZ + j], m);
    Cm[l * H_SZ + k] = m * W2[l * H_SZ + k];
}

// ---------------------------------------------------------------------------
// Generic fp32 WMMA GEMM: out = epilogue(A @ B + bias)
//   ACONCAT: A row m is [xs[m,0:66], t[m], 0...]  (K=67, zero padded)
//   EPI==0 : out0 = z                       (bias may be null)
//   EPI==1 : out0 = silu(z), out1 = silu'(z)
// ---------------------------------------------------------------------------
template<bool ACONCAT, int EPI>
__global__ __launch_bounds__(NT)
void gemm_wmma(const float* __restrict__ A, const float* __restrict__ xs,
               const float* __restrict__ tvec,
               const float* __restrict__ Bm, const float* __restrict__ bias,
               float* __restrict__ out0, float* __restrict__ out1,
               int M, int N, int K)
{
    __shared__ float As[BM][BK + 4];   // row stride 36 dwords
    __shared__ float Bt[BN][BK + 4];   // B tile TRANSPOSED: Bt[n][k] = B[k][n]

    const int tid  = threadIdx.x;
    const int wave = tid >> 5;
    const int lane = tid & 31;
    const int wm   = wave >> 1;          // 0..1
    const int wn   = wave & 1;           // 0..1

    const int m0 = blockIdx.y * BM;
    const int n0 = blockIdx.x * BN;

    v8f acc[2][2] = {};

    const int row16 = lane & 15;         // A row within 16 / B column within 16
    const int ksub  = (lane >> 4) << 1;  // lanes 0-15 -> K{0,1}; 16-31 -> K{2,3}

    for (int k0 = 0; k0 < K; k0 += BK) {
        // ---- stage A tile (BM x BK), global-coalesced over kk ----
        #pragma unroll
        for (int i = 0; i < (BM * BK) / NT; ++i) {
            int idx = tid + i * NT;
            int mm = idx >> 5;            // / BK
            int kk = idx & (BK - 1);
            int gk = k0 + kk;
            float v = 0.0f;
            if (ACONCAT) {
                int gm = m0 + mm;
                if (gk < D_SZ)       v = xs[gm * D_SZ + gk];
                else if (gk == D_SZ) v = tvec[gm];
            } else {
                if (gk < K) v = A[(size_t)(m0 + mm) * K + gk];
            }
            As[mm][kk] = v;
        }
        // ---- stage B tile transposed (Bt[n][k]), global-coalesced over nn ----
        #pragma unroll
        for (int i = 0; i < (BK * BN) / NT; ++i) {
            int idx = tid + i * NT;
            int kk = idx / BN;
            int nn = idx % BN;
            int gk = k0 + kk;
            int gn = n0 + nn;
            float v = 0.0f;
            if (gk < K && gn < N) v = Bm[(size_t)gk * N + gn];
            Bt[nn][kk] = v;
        }
        __syncthreads();

        // ---- 4 x v_wmma_f32_16x16x4_f32 per 4-K step; all frags are b64 ----
        #pragma unroll
        for (int kk = 0; kk < BK; kk += 4) {
            v2f a0 = *(const v2f*)&As[wm * 32 +      row16][kk + ksub];
            v2f a1 = *(const v2f*)&As[wm * 32 + 16 + row16][kk + ksub];
            v2f b0 = *(const v2f*)&Bt[wn * 32 +      row16][kk + ksub];
            v2f b1 = *(const v2f*)&Bt[wn * 32 + 16 + row16][kk + ksub];
            acc[0][0] = wmma4(a0, b0, acc[0][0]);
            acc[0][1] = wmma4(a0, b1, acc[0][1]);
            acc[1][0] = wmma4(a1, b0, acc[1][0]);
            acc[1][1] = wmma4(a1, b1, acc[1][1]);
        }
        __syncthreads();
    }

    // ---- epilogue: C/D layout: VGPR r -> row r + 8*(lane>=16), col lane%16 ----
    #pragma unroll
    for (int mt = 0; mt < 2; ++mt) {
        const int mbase = m0 + wm * 32 + mt * 16 + ((lane >> 4) << 3);
        #pragma unroll
        for (int nt = 0; nt < 2; ++nt) {
            const int nc = n0 + wn * 32 + nt * 16 + row16;
            if (nc >= N) continue;
            const float bi = bias ? bias[nc] : 0.0f;
            #pragma unroll
            for (int r = 0; r < 8; ++r) {
                const int gm = mbase + r;
                float z = acc[mt][nt][r] + bi;
                if (EPI == 0) {
                    out0[(size_t)gm * N + nc] = z;
                } else {
                    // silu(z) = z*sig ; silu'(z) = sig*(1 + z*(1-sig))
                    float sg = 1.0f / (1.0f + __expf(-z));
                    out0[(size_t)gm * N + nc] = z * sg;
                    out1[(size_t)gm * N + nc] = sg * (1.0f + z * (1.0f - sg));
                }
            }
        }
    }
}

// ---------------------------------------------------------------------------
// out[b] = -sum_k u[b,k] * s2[b,k]     (one 128-thread block per batch row)
// ---------------------------------------------------------------------------
__global__ __launch_bounds__(128)
void dot_div_kernel(const float* __restrict__ u, const float* __restrict__ s2,
                    float* __restrict__ out)
{
    int b = blockIdx.x;
    int tid = threadIdx.x;
    float sum = 0.0f;
    #pragma unroll
    for (int k = tid; k < H_SZ; k += 128)
        sum += u[(size_t)b * H_SZ + k] * s2[(size_t)b * H_SZ + k];
    #pragma unroll
    for (int off = 16; off > 0; off >>= 1)
        sum += __shfl_xor(sum, off, 32);
    __shared__ float red[4];
    if ((tid & 31) == 0) red[tid >> 5] = sum;
    __syncthreads();
    if (tid == 0) out[b] = -(red[0] + red[1] + red[2] + red[3]);
}

// ---------------------------------------------------------------------------
extern "C" void kernel_launch(void* const* d_in, const int* in_sizes, int n_in,
                              void* d_out, int out_size, void* d_ws, size_t ws_size,
                              hipStream_t stream) {
    const float* xs = (const float*)d_in[0];   // [B, D]
    const float* t  = (const float*)d_in[1];   // [B, 1]
    const float* W1 = (const float*)d_in[2];   // [D+1, H]
    const float* b1 = (const float*)d_in[3];   // [H]
    const float* W2 = (const float*)d_in[4];   // [H, H]
    const float* b2 = (const float*)d_in[5];   // [H]
    const float* W3 = (const float*)d_in[6];   // [H, D]
    const float* b3 = (const float*)d_in[7];   // [D]

    float* dxs  = (float*)d_out;                        // [B, D]
    float* ndiv = (float*)d_out + (size_t)B_SZ * D_SZ;  // [B, 1]

    // workspace layout (floats); a1 slot is reused for u once a1 is consumed
    float* ws = (float*)d_ws;
    const size_t act = (size_t)B_SZ * H_SZ;   // 2M floats each
    float* a1u = ws;              // a1, later u = s1 @ C
    float* s1  = ws + act;
    float* a2  = ws + 2 * act;
    float* s2  = ws + 3 * act;
    float* Cm  = ws + 4 * act;    // [H, H]

    dim3 blk(NT);
    dim3 gridH(H_SZ / BN, B_SZ / BM);                 // (8, 64)
    dim3 gridD((D_SZ + BN - 1) / BN, B_SZ / BM);      // (2, 64)

    // 0) batch-independent trace matrix C = W2 .* (W1[:D]^T W3^T)
    precompute_C<<<(H_SZ * H_SZ) / 256, 256, 0, stream>>>(W1, W2, W3, Cm);

    // 1) a1,s1 = silu,silu'( [xs,t] @ W1 + b1 )   (K = 67, zero padded)
    gemm_wmma<true, 1><<<gridH, blk, 0, stream>>>(
        nullptr, xs, t, W1, b1, a1u, s1, B_SZ, H_SZ, D_SZ + 1);

    // 2) a2,s2 = silu,silu'( a1 @ W2 + b2 )
    gemm_wmma<false, 1><<<gridH, blk, 0, stream>>>(
        a1u, nullptr, nullptr, W2, b2, a2, s2, B_SZ, H_SZ, H_SZ);

    // 3) u = s1 @ C        (overwrites a1 slot; a1 already consumed)
    gemm_wmma<false, 0><<<gridH, blk, 0, stream>>>(
        s1, nullptr, nullptr, Cm, nullptr, a1u, nullptr, B_SZ, H_SZ, H_SZ);

    // 4) dxs = a2 @ W3 + b3
    gemm_wmma<false, 0><<<gridD, blk, 0, stream>>>(
        a2, nullptr, nullptr, W3, b3, dxs, nullptr, B_SZ, D_SZ, H_SZ);

    // 5) -div[b] = -sum_k u[b,k] * s2[b,k]
    dot_div_kernel<<<B_SZ, 128, 0, stream>>>(a1u, s2, ndiv);
}